// CannyEdgeLayer_62388694941834
// MI455X (gfx1250) — compile-verified
//
#include <hip/hip_runtime.h>
#include <stdint.h>

#define IMG_H 1024
#define IMG_W 1024
#define LOW_T  0.1f
#define HIGH_T 0.3f
#define TAN225 0.41421356f
#define TAN675 2.4142137f

// CDNA5 async DMA: global -> LDS, tracked by ASYNCcnt (gfx1250-specific path)
__device__ __forceinline__ void async_ld_f32(const float* lds_dst, const float* gsrc)
{
    uint32_t lds_off = (uint32_t)(uintptr_t)lds_dst;
    uint64_t gaddr   = (uint64_t)(uintptr_t)gsrc;
    asm volatile("global_load_async_to_lds_b32 %0, %1, off"
                 :: "v"(lds_off), "v"(gaddr) : "memory");
}

// ---------------------------------------------------------------------------
// Kernel 1: fused blur + sobel + magnitude + NMS + double threshold.
// 16x16 output tile per 256-thread block. Input tile 22x22 (halo 3) staged
// into LDS with async global->LDS loads. Interior blocks (the vast majority)
// take a branch-free unconditional-load path; only image-border blocks pay
// for zero-fill + per-element bounds checks.
// Writes 1 byte per pixel: 2 = strong, 1 = weak, 0 = none.
// ---------------------------------------------------------------------------
__global__ __launch_bounds__(256)
void canny_front(const float* __restrict__ in, unsigned char* __restrict__ cls)
{
    __shared__ float s_in[22][23];   // input tile, halo 3
    __shared__ float s_bl[20][21];   // blurred, halo 2
    __shared__ float s_mag[18][19];  // gradient magnitude, halo 1

    const int tx = threadIdx.x, ty = threadIdx.y;
    const int tid = ty * 16 + tx;
    const int bi0 = blockIdx.y * 16 - 3;           // global row of s_in[0][*]
    const int bj0 = blockIdx.x * 16 - 3;           // global col of s_in[*][0]
    const float* img = in + ((size_t)blockIdx.z << 20);

    // uniform per block -> divergence-free, __syncthreads inside is legal
    const bool interior = (bi0 >= 0) & (bj0 >= 0) &
                          (bi0 + 22 <= IMG_H) & (bj0 + 22 <= IMG_W);

    if (interior) {
        // 484 elements, 256 threads: exactly 2 rounds, no bounds checks
        {
            const int t = tid;
            const int i = t / 22, j = t - i * 22;
            async_ld_f32(&s_in[i][j], img + (size_t)(bi0 + i) * IMG_W + (bj0 + j));
        }
        {
            const int t = tid + 256;
            if (t < 22 * 22) {
                const int i = t / 22, j = t - i * 22;
                async_ld_f32(&s_in[i][j], img + (size_t)(bi0 + i) * IMG_W + (bj0 + j));
            }
        }
    } else {
        // image-border blocks: zero halo, then load only in-bounds pixels
        float* sp = &s_in[0][0];
        for (int t = tid; t < 22 * 23; t += 256) sp[t] = 0.0f;
        __syncthreads();
        #pragma unroll
        for (int rep = 0; rep < 2; ++rep) {
            const int t = tid + rep * 256;
            if (t < 22 * 22) {
                const int i = t / 22, j = t - i * 22;
                const int gi = bi0 + i, gj = bj0 + j;
                if ((unsigned)gi < (unsigned)IMG_H && (unsigned)gj < (unsigned)IMG_W)
                    async_ld_f32(&s_in[i][j], img + (size_t)gi * IMG_W + gj);
            }
        }
    }
    asm volatile("s_wait_asynccnt 0" ::: "memory");
    __syncthreads();

    // 3x3 Gaussian blur (cross-correlation, weights /16): 400 outputs
    #pragma unroll
    for (int rep = 0; rep < 2; ++rep) {
        const int t = tid + rep * 256;
        if (t < 20 * 20) {
            const int i = t / 20, j = t - i * 20;   // s_in center = (i+1, j+1)
            float a00 = s_in[i][j],   a01 = s_in[i][j+1],   a02 = s_in[i][j+2];
            float a10 = s_in[i+1][j], a11 = s_in[i+1][j+1], a12 = s_in[i+1][j+2];
            float a20 = s_in[i+2][j], a21 = s_in[i+2][j+1], a22 = s_in[i+2][j+2];
            s_bl[i][j] = ((a00 + a02 + a20 + a22)
                       + 2.0f * (a01 + a10 + a12 + a21)
                       + 4.0f * a11) * 0.0625f;
        }
    }
    __syncthreads();

    // Sobel magnitude: 324 outputs
    #pragma unroll
    for (int rep = 0; rep < 2; ++rep) {
        const int t = tid + rep * 256;
        if (t < 18 * 18) {
            const int i = t / 18, j = t - i * 18;   // s_bl center = (i+1, j+1)
            float gx = (s_bl[i][j+2]   - s_bl[i][j])
                     + 2.0f * (s_bl[i+1][j+2] - s_bl[i+1][j])
                     + (s_bl[i+2][j+2] - s_bl[i+2][j]);
            float gy = (s_bl[i+2][j]   - s_bl[i][j])
                     + 2.0f * (s_bl[i+2][j+1] - s_bl[i][j+1])
                     + (s_bl[i+2][j+2] - s_bl[i][j+2]);
            s_mag[i][j] = sqrtf(gx * gx + gy * gy);
        }
    }
    __syncthreads();

    // per-pixel: direction quantization + NMS + double threshold
    {
        const int i = ty + 2, j = tx + 2;           // s_bl coords of this pixel
        float gx = (s_bl[i-1][j+1] - s_bl[i-1][j-1])
                 + 2.0f * (s_bl[i][j+1] - s_bl[i][j-1])
                 + (s_bl[i+1][j+1] - s_bl[i+1][j-1]);
        float gy = (s_bl[i+1][j-1] - s_bl[i-1][j-1])
                 + 2.0f * (s_bl[i+1][j] - s_bl[i-1][j])
                 + (s_bl[i+1][j+1] - s_bl[i-1][j+1]);

        const float ax = fabsf(gx), ay = fabsf(gy);
        const bool b0 = ay < TAN225 * ax;                       // ~horizontal grad
        const bool b2 = ay >= TAN675 * ax;                      // ~vertical grad
        const bool b1 = (!b0) && (!b2) && (gx * gy > 0.0f);     // 45 deg

        int d1i, d1j;
        if (b0)      { d1i = 0;  d1j = 1;  }
        else if (b1) { d1i = -1; d1j = 1;  }
        else if (b2) { d1i = -1; d1j = 0;  }
        else         { d1i = -1; d1j = -1; }

        const float m  = s_mag[ty + 1][tx + 1];
        const float n1 = s_mag[ty + 1 + d1i][tx + 1 + d1j];
        const float n2 = s_mag[ty + 1 - d1i][tx + 1 - d1j];
        const float sup = (m >= n1 && m >= n2) ? m : 0.0f;

        unsigned char c = (sup >= HIGH_T) ? 2 : ((sup >= LOW_T) ? 1 : 0);
        const int gi = blockIdx.y * 16 + ty;
        const int gj = blockIdx.x * 16 + tx;
        cls[((size_t)blockIdx.z << 20) + (size_t)gi * IMG_W + gj] = c;
    }
}

// ---------------------------------------------------------------------------
// Kernel 2: all 16 hysteresis iterations fused, exact.
// 64-thread block, tile 64x64 (output 32x32, halo 16): after 16 iterations a
// pixel depends only on initial state within Chebyshev radius 16, so the
// 32x32 interior is bit-exact vs. 16 global dilations. Each thread owns one
// row as a 64-bit mask; each iteration is ~6 bitwise ops + one LDS b64
// exchange + barrier.
// ---------------------------------------------------------------------------
__device__ __forceinline__ uint32_t nib4(uint32_t x) {
    // x has bits only at positions 0,8,16,24 -> compact into bits 0..3
    return (x | (x >> 7) | (x >> 14) | (x >> 21)) & 0xFu;
}

__global__ __launch_bounds__(64)
void canny_hyst(const unsigned char* __restrict__ cls, float* __restrict__ out)
{
    __shared__ unsigned long long rows[66];    // [0] and [65] stay zero

    const int r   = threadIdx.x;               // 0..63 : tile row
    const int gi  = blockIdx.y * 32 - 16 + r;  // global row
    const int gj0 = blockIdx.x * 32 - 16;      // global col of tile col 0
    const unsigned char* base = cls + ((size_t)blockIdx.z << 20);

    unsigned long long weak = 0ull, strong = 0ull;
    if ((unsigned)gi < (unsigned)IMG_H) {
        if (gj0 >= 0 && gj0 + 64 <= IMG_W) {
            // fast interior path: 4x uint4 (16B, aligned since gj0 % 16 == 0)
            const uint4* p = (const uint4*)(base + (size_t)gi * IMG_W + gj0);
            #pragma unroll
            for (int q = 0; q < 4; ++q) {
                uint4 v = p[q];
                uint32_t w[4] = { v.x, v.y, v.z, v.w };
                #pragma unroll
                for (int k = 0; k < 4; ++k) {
                    const int sh = (q * 4 + k) * 4;
                    weak   |= (unsigned long long)nib4(w[k] & 0x01010101u) << sh;
                    strong |= (unsigned long long)nib4((w[k] >> 1) & 0x01010101u) << sh;
                }
            }
        } else {
            // image-edge path: per-byte with bounds check (zeros outside)
            for (int j = 0; j < 64; ++j) {
                const int gj = gj0 + j;
                unsigned c = ((unsigned)gj < (unsigned)IMG_W)
                           ? (unsigned)base[(size_t)gi * IMG_W + gj] : 0u;
                weak   |= (unsigned long long)(c & 1u) << j;
                strong |= (unsigned long long)((c >> 1) & 1u) << j;
            }
        }
    }

    if (r == 0) { rows[0] = 0ull; rows[65] = 0ull; }
    unsigned long long edge = strong;

    #pragma unroll 4
    for (int it = 0; it < 16; ++it) {
        rows[r + 1] = edge;
        __syncthreads();
        unsigned long long h = rows[r] | rows[r + 1] | rows[r + 2];
        h |= (h << 1) | (h >> 1);              // 3x3 dilation of edges
        edge |= (weak & h);                    // grow onto weak pixels
        __syncthreads();
    }

    // coalesced output of the exact 32x32 interior
    rows[r + 1] = edge;
    __syncthreads();

    float* obase = out + ((size_t)blockIdx.z << 20);
    const int oi0 = blockIdx.y * 32, oj0 = blockIdx.x * 32;
    for (int e = r; e < 32 * 32; e += 64) {
        const int oi = e >> 5, oj = e & 31;
        const unsigned long long row = rows[16 + oi + 1];
        const float val = ((row >> (16 + oj)) & 1ull) ? 1.0f : 0.0f;
        obase[(size_t)(oi0 + oi) * IMG_W + (oj0 + oj)] = val;
    }
}

// ---------------------------------------------------------------------------
extern "C" void kernel_launch(void* const* d_in, const int* in_sizes, int n_in,
                              void* d_out, int out_size, void* d_ws, size_t ws_size,
                              hipStream_t stream)
{
    (void)n_in; (void)out_size; (void)ws_size;
    const float* in = (const float*)d_in[0];
    float* out = (float*)d_out;
    unsigned char* cls = (unsigned char*)d_ws;   // 1 byte/pixel scratch

    const int B = in_sizes[0] / (IMG_H * IMG_W);

    dim3 g1(IMG_W / 16, IMG_H / 16, B), b1(16, 16, 1);
    canny_front<<<g1, b1, 0, stream>>>(in, cls);

    dim3 g2(IMG_W / 32, IMG_H / 32, B), b2(64, 1, 1);
    canny_hyst<<<g2, b2, 0, stream>>>(cls, out);
}